// MBGCN_59107339927714
// MI455X (gfx1250) — compile-verified
//
#include <hip/hip_runtime.h>

typedef __attribute__((ext_vector_type(2))) float v2f;
typedef __attribute__((ext_vector_type(8))) float v8f;

#define NUM_USERS 500000
#define NUM_ITEMS 100000
#define EMB 64
#define NB 3
#define BATCH 1048576
#define LAMB 0.5f

// ---------------------------------------------------------------------------
// Kernel 1: Q[t] = S[t] (NUM_ITEMS x EMB) @ M[t]^T (EMB x EMB), fp32 WMMA.
// One wave32 per 16x16 output tile; K=64 swept in 16 steps of 4 with
// V_WMMA_F32_16X16X4_F32.
//
// ISA layouts (CDNA5 ISA):
//   A 16x4 f32 : lane L<16 holds (M=L, K=kk..kk+1), lanes 16-31 hold K=kk+2..kk+3
//   B 4x16 f32 : lane L<16 holds (K=kk..kk+1, N=L), lanes 16-31 hold K=kk+2..kk+3
//   C/D 16x16  : VGPR v, lanes 0-15 -> (M=v, N=lane); lanes 16-31 -> (M=v+8)
// ---------------------------------------------------------------------------
__global__ void mbgcn_precompute_Q(const float* __restrict__ S,   // [NB][NUM_ITEMS][EMB]
                                   const float* __restrict__ M,   // [NB][EMB][EMB]
                                   float* __restrict__ Q)         // [NB][NUM_ITEMS][EMB]
{
    constexpr int ROW_TILES = NUM_ITEMS / 16;   // 6250 (exact)
    constexpr int COL_TILES = EMB / 16;         // 4
    const int wave = (blockIdx.x * blockDim.x + threadIdx.x) >> 5;
    const int lane = threadIdx.x & 31;
    const int total = NB * ROW_TILES * COL_TILES;
    if (wave >= total) return;

    const int t       = wave / (ROW_TILES * COL_TILES);
    const int rem     = wave % (ROW_TILES * COL_TILES);
    const int rowTile = rem / COL_TILES;
    const int colTile = rem % COL_TILES;

    const int lane16 = lane & 15;
    const int khalf  = (lane >> 4) * 2;         // lanes 16-31 hold K+2,K+3

    // A: rows of S[t] for this tile.  B[k][n] = M[t][colBase+n][k]  (Q = S * M^T)
    const float* __restrict__ Srow = S + ((size_t)t * NUM_ITEMS + (size_t)rowTile * 16 + lane16) * EMB;
    const float* __restrict__ Mrow = M + ((size_t)t * EMB       + (size_t)colTile * 16 + lane16) * EMB;

    v8f c = {};
#pragma unroll
    for (int kk = 0; kk < EMB; kk += 4) {
        v2f a = *(const v2f*)(Srow + kk + khalf);
        v2f b = *(const v2f*)(Mrow + kk + khalf);
        // (neg_a, A, neg_b, B, c_mod, C, reuse_a, reuse_b)
        c = __builtin_amdgcn_wmma_f32_16x16x4_f32(false, a, false, b,
                                                  (short)0, c, false, false);
    }

    const int mbase = (lane >> 4) * 8;
    float* __restrict__ Qbase = Q + ((size_t)t * NUM_ITEMS + (size_t)rowTile * 16) * EMB
                                  + (size_t)colTile * 16 + lane16;
#pragma unroll
    for (int v = 0; v < 8; ++v) {
        Qbase[(size_t)(mbase + v) * EMB] = c[v];
    }
}

// ---------------------------------------------------------------------------
// Kernel 2: memory-bound gather + dot.  Each half-wave (16 lanes x float4)
// owns one batch element so each 256-byte embedding row is one coalesced
// read.  out[b] = 0.5*dot(u,i) + 0.5 * sum_t w_t * dot(um[t,u], Q[t,i]).
// ---------------------------------------------------------------------------
__global__ void mbgcn_score(const int*   __restrict__ user_idx,
                            const int*   __restrict__ item_idx,
                            const float* __restrict__ user_latent,   // [NUM_USERS][EMB]
                            const float* __restrict__ item_latent,   // [NUM_ITEMS][EMB]
                            const float* __restrict__ user_mean_emb, // [NB][NUM_USERS][EMB]
                            const float* __restrict__ Q,             // [NB][NUM_ITEMS][EMB]
                            const float* __restrict__ alpha,         // [NB]
                            float*       __restrict__ out)           // [BATCH]
{
    const int lane        = threadIdx.x & 31;
    const int lane16      = lane & 15;
    const int half        = lane >> 4;
    const int waveInBlock = threadIdx.x >> 5;
    const size_t b = (size_t)blockIdx.x * 16 + (size_t)waveInBlock * 2 + half;

    const float a0 = alpha[0], a1 = alpha[1], a2 = alpha[2];
    const float inv = 1.0f / (a0 + a1 + a2);
    const float w0 = a0 * inv, w1 = a1 * inv, w2 = a2 * inv;

    const int u  = user_idx[b];
    const int it = item_idx[b];

    const float4 uu = ((const float4*)(user_latent + (size_t)u  * EMB))[lane16];
    const float4 ii = ((const float4*)(item_latent + (size_t)it * EMB))[lane16];
    float ucf = uu.x * ii.x + uu.y * ii.y + uu.z * ii.z + uu.w * ii.w;

    float icf = 0.0f;
    const float wv[NB] = {w0, w1, w2};
#pragma unroll
    for (int t = 0; t < NB; ++t) {
        const float4 um = ((const float4*)(user_mean_emb + ((size_t)t * NUM_USERS + u ) * EMB))[lane16];
        const float4 qv = ((const float4*)(Q             + ((size_t)t * NUM_ITEMS + it) * EMB))[lane16];
        icf += wv[t] * (um.x * qv.x + um.y * qv.y + um.z * qv.z + um.w * qv.w);
    }

    // Reduce across the 16 lanes of this half-wave (xor masks <= 8 stay
    // inside the 16-lane group on wave32).
#pragma unroll
    for (int off = 8; off >= 1; off >>= 1) {
        ucf += __shfl_xor(ucf, off, 16);
        icf += __shfl_xor(icf, off, 16);
    }

    if (lane16 == 0) {
        out[b] = LAMB * ucf + (1.0f - LAMB) * icf;
    }
}

// ---------------------------------------------------------------------------
// d_in order: user_idx, item_idx, user_latent, item_latent, s_item_list,
//             user_mean_emb, M_t, behavior_alpha
// d_ws: needs NB*NUM_ITEMS*EMB*4 = 76.8 MB for the transformed item table Q.
// ---------------------------------------------------------------------------
extern "C" void kernel_launch(void* const* d_in, const int* in_sizes, int n_in,
                              void* d_out, int out_size, void* d_ws, size_t ws_size,
                              hipStream_t stream) {
    const int*   user_idx      = (const int*)  d_in[0];
    const int*   item_idx      = (const int*)  d_in[1];
    const float* user_latent   = (const float*)d_in[2];
    const float* item_latent   = (const float*)d_in[3];
    const float* s_item_list   = (const float*)d_in[4];
    const float* user_mean_emb = (const float*)d_in[5];
    const float* M_t           = (const float*)d_in[6];
    const float* alpha         = (const float*)d_in[7];
    float*       out           = (float*)d_out;
    float*       Q             = (float*)d_ws;   // NB*NUM_ITEMS*EMB floats

    // Kernel 1: WMMA GEMM for Q[t] = S[t] @ M[t]^T
    const int totalTiles = NB * (NUM_ITEMS / 16) * (EMB / 16);   // 75000 waves
    const int wavesPerBlock = 256 / 32;
    const int blocks1 = (totalTiles + wavesPerBlock - 1) / wavesPerBlock;
    mbgcn_precompute_Q<<<blocks1, 256, 0, stream>>>(s_item_list, M_t, Q);

    // Kernel 2: gather + dot over the batch (16 elements per 256-thread block)
    const int blocks2 = BATCH / 16;                              // 65536
    mbgcn_score<<<blocks2, 256, 0, stream>>>(user_idx, item_idx, user_latent,
                                             item_latent, user_mean_emb, Q,
                                             alpha, out);
}